// ReprojectionMultiRigModel_68839735820964
// MI455X (gfx1250) — compile-verified
//
#include <hip/hip_runtime.h>

// ReprojectionMultiRigModel for MI455X (gfx1250, wave32).
// Memory-bound gather/elementwise kernel (~32 B/elem streamed HBM traffic;
// pose/point tables are 280 KB + 6 MB -> L2/WGP$ resident on a 192 MB L2).
// 2 elements/thread for b128-wide NT streaming loads/stores; LDS staging for
// the tiny rel-pose/intrinsics tables; NT temporal hints so single-touch
// streams don't pollute L2.

typedef float f32x2 __attribute__((ext_vector_type(2)));
typedef float f32x4 __attribute__((ext_vector_type(4)));
typedef int   i32x2 __attribute__((ext_vector_type(2)));
typedef int   i32x4 __attribute__((ext_vector_type(4)));

#define NCAMS 8
#define NPOS  8

struct F3 { float x, y, z; };

__device__ __forceinline__ F3 cross3(const F3 a, const F3 b) {
    return F3{ a.y * b.z - a.z * b.y,
               a.z * b.x - a.x * b.z,
               a.x * b.y - a.y * b.x };
}

__device__ __forceinline__ F3 add3(const F3 a, const F3 b) {
    return F3{ a.x + b.x, a.y + b.y, a.z + b.z };
}

// Rotate p by unit quaternion (v, w), pypose [x,y,z,w] convention:
// p + 2*(w*cross(v,p) + cross(v, cross(v,p)))
__device__ __forceinline__ F3 quat_rotate(const F3 v, const float w, const F3 p) {
    F3 uv  = cross3(v, p);
    F3 uuv = cross3(v, uv);
    return F3{ p.x + 2.0f * (w * uv.x + uuv.x),
               p.y + 2.0f * (w * uv.y + uuv.y),
               p.z + 2.0f * (w * uv.z + uuv.z) };
}

// One observation: indices + observed pixel -> residual.
__device__ __forceinline__ f32x2 process_one(int g, int m, int pt, int cam,
                                             float obs_x, float obs_y,
                                             const float* __restrict__ points_3d,
                                             const float* __restrict__ ref_poses,
                                             const float* s_rel,
                                             const float* s_intr,
                                             const float* s_pp)
{
    // Gather ref pose: 280 KB table, regular-temporal -> L2/WGP$ resident.
    const float* __restrict__ ref = ref_poses + 7 * g;
    const F3    ref_t  = { ref[0], ref[1], ref[2] };
    const F3    ref_qv = { ref[3], ref[4], ref[5] };
    const float ref_qw = ref[6];

    // Rel pose from LDS.
    const float* rel = s_rel + 7 * m;
    const F3    rel_t  = { rel[0], rel[1], rel[2] };
    const F3    rel_qv = { rel[3], rel[4], rel[5] };
    const float rel_qw = rel[6];

    // t = rel_t + R(rel_q) * ref_t
    const F3 tvec = add3(rel_t, quat_rotate(rel_qv, rel_qw, ref_t));

    // q = rel_q (x) ref_q   (Hamilton product, [x,y,z,w] storage)
    const float qw = rel_qw * ref_qw
                   - (rel_qv.x * ref_qv.x + rel_qv.y * ref_qv.y + rel_qv.z * ref_qv.z);
    const F3 cxyz = cross3(rel_qv, ref_qv);
    const F3 qv = { rel_qw * ref_qv.x + ref_qw * rel_qv.x + cxyz.x,
                    rel_qw * ref_qv.y + ref_qw * rel_qv.y + cxyz.y,
                    rel_qw * ref_qv.z + ref_qw * rel_qv.z + cxyz.z };

    // Gather 3D point: 6 MB table, regular-temporal -> L2 resident.
    const float* __restrict__ p3 = points_3d + 3 * pt;
    const F3 P = { p3[0], p3[1], p3[2] };

    // Camera-frame point and pinhole projection.
    const F3 pc = add3(quat_rotate(qv, qw, P), tvec);
    const float invz = 1.0f / pc.z;

    const float fx = s_intr[2 * cam + 0];
    const float fy = s_intr[2 * cam + 1];
    const float cx = s_pp[2 * cam + 0];
    const float cy = s_pp[2 * cam + 1];

    f32x2 r;
    r.x = fx * (pc.x * invz) + cx - obs_x;
    r.y = fy * (pc.y * invz) + cy - obs_y;
    return r;
}

__global__ __launch_bounds__(256)
void reproj_multirig_kernel(const float* __restrict__ points_2d,
                            const int*   __restrict__ camera_indices,
                            const int*   __restrict__ grouping,
                            const int*   __restrict__ point_indices,
                            const float* __restrict__ camera_pps,
                            const float* __restrict__ intrs,
                            const float* __restrict__ points_3d,
                            const float* __restrict__ ref_poses,
                            const float* __restrict__ rel_poses,
                            float*       __restrict__ out,
                            int n)
{
    // Stage the tiny hot tables in LDS (352 B of the 320 KB/WGP):
    // rel_poses: 8 poses x 7 f32, intrs/pps: 8 cams x 2 f32 each.
    __shared__ float s_rel[NPOS * 7];
    __shared__ float s_intr[NCAMS * 2];
    __shared__ float s_pp[NCAMS * 2];

    const int t = threadIdx.x;
    if (t < NPOS * 7)           s_rel[t]       = rel_poses[t];
    else if (t < NPOS * 7 + 16) s_intr[t - 56] = intrs[t - 56];
    else if (t < NPOS * 7 + 32) s_pp[t - 72]   = camera_pps[t - 72];
    __syncthreads();

    // Two observations per thread: wide (b128) coalesced NT streaming.
    const int i0 = (blockIdx.x * 256 + t) * 2;

    if (i0 + 1 < n) {
        // 16B-aligned vector loads (i0 is even).
        const i32x4 gm2  = __builtin_nontemporal_load((const i32x4*)(grouping      + 2 * i0));
        const f32x4 obs2 = __builtin_nontemporal_load((const f32x4*)(points_2d    + 2 * i0));
        const i32x2 pt2  = __builtin_nontemporal_load((const i32x2*)(point_indices +     i0));
        const i32x2 cam2 = __builtin_nontemporal_load((const i32x2*)(camera_indices +   i0));

        const f32x2 r0 = process_one(gm2.x, gm2.y, pt2.x, cam2.x, obs2.x, obs2.y,
                                     points_3d, ref_poses, s_rel, s_intr, s_pp);
        const f32x2 r1 = process_one(gm2.z, gm2.w, pt2.y, cam2.y, obs2.z, obs2.w,
                                     points_3d, ref_poses, s_rel, s_intr, s_pp);

        f32x4 r;
        r.x = r0.x; r.y = r0.y; r.z = r1.x; r.w = r1.y;
        __builtin_nontemporal_store(r, (f32x4*)(out + 2 * i0));
    } else if (i0 < n) {
        // Odd tail element.
        const int g   = grouping[2 * i0 + 0];
        const int m   = grouping[2 * i0 + 1];
        const int pt  = point_indices[i0];
        const int cam = camera_indices[i0];
        const float ox = points_2d[2 * i0 + 0];
        const float oy = points_2d[2 * i0 + 1];

        const f32x2 r = process_one(g, m, pt, cam, ox, oy,
                                    points_3d, ref_poses, s_rel, s_intr, s_pp);
        __builtin_nontemporal_store(r, (f32x2*)(out + 2 * i0));
    }
}

extern "C" void kernel_launch(void* const* d_in, const int* in_sizes, int n_in,
                              void* d_out, int out_size, void* d_ws, size_t ws_size,
                              hipStream_t stream) {
    // setup_inputs() order:
    // 0 points_2d (N,2) f32        1 camera_indices (N,) i32
    // 2 grouping_indices (N,2) i32 3 point_indices (N,) i32
    // 4 camera_pps (8,2) f32       5 intrs (8,2) f32
    // 6 points_3d (NUM_PTS,3) f32
    // 7 ref_poses (NUM_GROUPS,7) f32   8 rel_poses (8,7) f32
    const float* points_2d      = (const float*)d_in[0];
    const int*   camera_indices = (const int*)  d_in[1];
    const int*   grouping       = (const int*)  d_in[2];
    const int*   point_indices  = (const int*)  d_in[3];
    const float* camera_pps     = (const float*)d_in[4];
    const float* intrs          = (const float*)d_in[5];
    const float* points_3d      = (const float*)d_in[6];
    const float* ref_poses      = (const float*)d_in[7];
    const float* rel_poses      = (const float*)d_in[8];
    float*       out            = (float*)d_out;

    const int n = in_sizes[1];                 // N observations
    const int elems_per_block = 256 * 2;       // 2 observations per thread
    const int blocks = (n + elems_per_block - 1) / elems_per_block;

    hipLaunchKernelGGL(reproj_multirig_kernel, dim3(blocks), dim3(256), 0, stream,
                       points_2d, camera_indices, grouping, point_indices,
                       camera_pps, intrs, points_3d, ref_poses, rel_poses,
                       out, n);
}